// VCN_1116691497113
// MI455X (gfx1250) — compile-verified
//
#include <hip/hip_runtime.h>
#include <math.h>

// Problem constants (from the reference)
#define MDISP 4
#define NU 9
#define NV 9
#define NUV 81          // 9*9 displacement plane
#define NH 96
#define NW 96
#define NHW (NH * NW)   // 9216
#define NB 64

#define TPB 128                     // pixels per block (one LDS column each)
#define TILES_PER_B (NHW / TPB)     // 72
#define ROWBYTES (TPB * 4)          // 512 B = one wave-wide b128 async op
#define WAVES (TPB / 32)            // 4

__global__ __launch_bounds__(TPB) void vcn_flow_entropy_kernel(
    const float* __restrict__ x, float* __restrict__ out) {
  // 81 rows x 128 pixels, f32 -> 41472 bytes of LDS (~7 blocks / 320KB WGP)
  __shared__ float tile[NUV * TPB];

  const int blk = blockIdx.x;
  const int b   = blk / TILES_PER_B;
  const int t   = blk % TILES_PER_B;
  const int hw0 = t * TPB;
  const int c   = threadIdx.x;        // column (pixel) owned by this thread

  // ---------------------------------------------------------------------
  // Stage the 81x128 tile with CDNA5 async global->LDS DMA (ASYNCcnt path).
  // One row = 512 B = one wave-wide b128 async transfer (32 lanes x 16 B).
  // Input is streamed exactly once -> non-temporal load hint.
  // ---------------------------------------------------------------------
  const float* gbase = x + (size_t)b * (size_t)(NUV * NHW) + (size_t)hw0;
  const unsigned lds0 = (unsigned)(size_t)(&tile[0]);  // low 32 bits = LDS offset

  {
    const int wave = threadIdx.x >> 5;
    const int lane = threadIdx.x & 31;
    for (int r = wave; r < NUV; r += WAVES) {
      unsigned goff = (unsigned)(r * (NHW * 4)) + (unsigned)(lane * 16);
      unsigned loff = lds0 + (unsigned)(r * ROWBYTES) + (unsigned)(lane * 16);
      asm volatile("global_load_async_to_lds_b128 %0, %1, %2 offset:0 th:TH_LOAD_NT"
                   :: "v"(loff), "v"(goff), "s"(gbase) : "memory");
    }
    asm volatile("s_wait_asynccnt 0" ::: "memory");
  }
  __syncthreads();

  // ---------------------------------------------------------------------
  // Pass 1: max + argmax over the 81 displacement values of this pixel.
  // Column accesses are lane-consecutive 4B -> bank-conflict free.
  // ---------------------------------------------------------------------
  float m = -3.402823466e+38f;
  int best = 0;
#pragma unroll 9
  for (int r = 0; r < NUV; ++r) {
    float v = tile[r * TPB + c];
    if (v > m) { m = v; best = r; }
  }
  const int ub = best / NV;
  const int vb = best - ub * NV;

  // 9-bit window-membership masks for the 7x7 Chebyshev window around the
  // winner: bit u of mu is set iff |u - ub| <= 3 (clipped to [0,8]).
  const unsigned mu = (0x7Fu << ub) >> 3;
  const unsigned mv = (0x7Fu << vb) >> 3;

  // ---------------------------------------------------------------------
  // Pass 2 (single remaining data pass): with d = x - m, e = exp(d):
  //   S  = sum e            T  = sum e*d          (global softmax stats)
  //   St = sum_win e        Tt = sum_win e*d      (truncated stats)
  //   Sx = sum_win e*(u-4)  Sy = sum_win e*(v-4)  (flow numerators)
  // Entropy identity (matches reference up to ~3e-8 from the 1e-9 clip):
  //   -sum p ln p = ln(S) - T/S
  // ---------------------------------------------------------------------
  float S = 0.f, T = 0.f, St = 0.f, Tt = 0.f, Sx = 0.f, Sy = 0.f;
  for (int u = 0; u < NU; ++u) {
    const bool urow = (mu >> u) & 1u;
    const float fu = (float)(u - MDISP);
#pragma unroll
    for (int vv = 0; vv < NV; ++vv) {
      const int r = u * NV + vv;
      float v = tile[r * TPB + c];
      float d = v - m;
      float e = __expf(d);
      S += e;
      T  = fmaf(e, d, T);
      const bool in = urow && ((mv >> vv) & 1u);
      float w = in ? e : 0.0f;
      St += w;
      Tt = fmaf(w, d, Tt);
      Sx = fmaf(w, fu, Sx);
      Sy = fmaf(w, (float)(vv - MDISP), Sy);
    }
  }

  const float invS  = 1.0f / S;
  const float invSt = 1.0f / St;
  const float EL = __logf(St) - Tt * invSt;   // truncated entropy (nats)
  const float EG = __logf(S)  - T  * invS;    // global entropy (nats)

  // ---------------------------------------------------------------------
  // Output [b, 4, h, w]: outx, outy, local_ent/ln(49), global_ent/ln(81)
  // ---------------------------------------------------------------------
  const int hw = hw0 + c;
  float* ob = out + (size_t)b * (size_t)(4 * NHW) + (size_t)hw;
  ob[0]       = Sx * invSt;
  ob[NHW]     = Sy * invSt;
  ob[2 * NHW] = EL * 0.25694918f;   // 1 / ln(49)
  ob[3 * NHW] = EG * 0.22755998f;   // 1 / ln(81)
}

extern "C" void kernel_launch(void* const* d_in, const int* in_sizes, int n_in,
                              void* d_out, int out_size, void* d_ws, size_t ws_size,
                              hipStream_t stream) {
  (void)in_sizes; (void)n_in; (void)out_size; (void)d_ws; (void)ws_size;
  const float* x = (const float*)d_in[0];
  float* out = (float*)d_out;
  const int grid = NB * TILES_PER_B;  // 64 * 72 = 4608 blocks
  vcn_flow_entropy_kernel<<<grid, TPB, 0, stream>>>(x, out);
}